// GCNBlock_81776177316092
// MI455X (gfx1250) — compile-verified
//
#include <hip/hip_runtime.h>
#include <math.h>

typedef __attribute__((ext_vector_type(2))) float v2f;
typedef __attribute__((ext_vector_type(8))) float v8f;
typedef __attribute__((ext_vector_type(4))) unsigned int v4u;
typedef __attribute__((ext_vector_type(4))) int v4i;
typedef __attribute__((ext_vector_type(8))) int v8i;

#define NB 64
#define V  256
#define F  64
#define C  64
#define ALPHA_W 0.1f

// d_out layout (floats): gcn [NB*V*C] | s [NB*V*V] | ajloss [1]
#define GCN_ELEMS (NB * V * C)   // 1048576
#define S_ELEMS   (NB * V * V)   // 4194304

// workspace layout (floats): d [NB*V] | y1 [NB*V*F] | y2 [NB*V*F]
#define WS_D  0
#define WS_Y1 (NB * V)
#define WS_Y2 (WS_Y1 + NB * V * F)

#if defined(__has_builtin)
#if __has_builtin(__builtin_amdgcn_tensor_load_to_lds) && __has_builtin(__builtin_amdgcn_s_wait_tensorcnt)
#define HAVE_TDM 1
#endif
#endif
#if __has_include(<hip/amd_detail/amd_gfx1250_TDM.h>)
#define TDM_SIX_ARGS 1
#endif

__global__ void zero_loss(float* loss) {
    if (threadIdx.x == 0) loss[0] = 0.0f;
}

// One block per batch n; thread j owns column j of the 256x256 score matrix.
__global__ __launch_bounds__(256) void graph_learn(const float* __restrict__ x,
                                                   const float* __restrict__ a,
                                                   float* __restrict__ s_out,
                                                   float* __restrict__ d_ws,
                                                   float* __restrict__ loss) {
    __shared__ float lx[V * F];   // 64 KB x-tile for this batch
    __shared__ float la[F];
    __shared__ float lred[256];

    const int n   = blockIdx.x;
    const int tid = threadIdx.x;
    const float* xb = x + (size_t)n * V * F;

#if defined(HAVE_TDM)
    // Stage x[n] (256x64 f32 = 64KB) into LDS with one Tensor Data Mover op.
    // D# built per CDNA5 ISA 8.3/8.4: 1 row of 16384 4-byte elements.
    if ((tid >> 5) == 0) {                      // wave 0 only (EXEC all-1s)
        const unsigned long long ga = (unsigned long long)xb;
        const unsigned lds_off = (unsigned)(unsigned long long)(void*)lx;
        v4u g0 = { 0x1u,                        // count=1, user descriptor
                   lds_off,                     // lds_addr [63:32]
                   (unsigned)(ga & 0xffffffffu),
                   (unsigned)(((ga >> 32) & 0x01ffffffull) | 0x80000000u) }; // type=2
        v8i g1 = { (int)0x00020000,             // wg_mask=0, data_size=2 (4B)
                   (int)0x40000000,             // tensor_dim0 lo16 = 16384 (<<16)
                   (int)0x00010000,             // tensor_dim0 hi=0, tensor_dim1=1
                   (int)0x40000000,             // tile_dim0 = 16384 (<<16)
                   (int)0x00000001,             // tile_dim1 = 1, tile_dim2 = 0
                   (int)16384,                  // tensor_dim0_stride lo32
                   0, 0 };
        v4i gz = { 0, 0, 0, 0 };
#if defined(TDM_SIX_ARGS)
        v8i gz8 = { 0, 0, 0, 0, 0, 0, 0, 0 };
        __builtin_amdgcn_tensor_load_to_lds(g0, g1, gz, gz, gz8, 0);
#else
        __builtin_amdgcn_tensor_load_to_lds(g0, g1, gz, gz, 0);
#endif
        __builtin_amdgcn_s_wait_tensorcnt(0);
    }
#else
    // Fallback: cooperative float4 copy
    {
        const float4* xb4 = (const float4*)xb;
        float4* lx4 = (float4*)lx;
        for (int i = tid; i < (V * F) / 4; i += 256) lx4[i] = xb4[i];
    }
#endif
    if (tid < F) la[tid] = a[tid];
    __syncthreads();

    const int j = tid;
    float xj[F];
#pragma unroll
    for (int f = 0; f < F; ++f) xj[f] = lx[j * F + f];

    float colsum = 0.0f, sum2 = 0.0f, sumd = 0.0f;
    float* scol = s_out + (size_t)n * V * V;   // s[n, i, j] at i*V + j

    for (int i = 0; i < V; ++i) {
        float score = 0.0f, dist2 = 0.0f;
#pragma unroll
        for (int f = 0; f < F; ++f) {
            float df = lx[i * F + f] - xj[f];      // broadcast LDS read
            score = fmaf(fabsf(df), la[f], score);
            dist2 = fmaf(df, df, dist2);
        }
        float t = __expf(-fmaxf(score, 0.0f));
        scol[i * V + j] = t;                       // unnormalized, fixed below
        colsum += t;
        sum2 = fmaf(t, t, sum2);
        sumd = fmaf(t, dist2, sumd);
    }

    float inv  = 1.0f / colsum;
    float dsum = 0.0f;
    for (int i = 0; i < V; ++i) {
        float sv = scol[i * V + j] * inv;
        scol[i * V + j] = sv;
        dsum += sv;
    }
    d_ws[n * V + j] = dsum;                        // degree (column sum of s)

    // ajloss = sum s^2 + ALPHA * sum s*dist2, folded per column
    float lcol = sum2 * inv * inv + ALPHA_W * sumd * inv;
    lred[tid] = lcol;
    __syncthreads();
    for (int off = 128; off > 0; off >>= 1) {
        if (tid < off) lred[tid] += lred[tid + off];
        __syncthreads();
    }
    if (tid == 0) atomicAdd(loss, lred[0]);
}

// y1 = l_t @ x ; y2 = (l_t .* l_t) @ x   with l_t[i,j] = -s[i,j] + (i==j ? d[j]-1 : 0)
// One wave per 16x16 tile; branch-free K-loop accumulates -s everywhere, the
// diagonal delta is applied to the output tile afterwards (once per row).
__global__ __launch_bounds__(128) void lap_gemm(const float* __restrict__ x,
                                                const float* __restrict__ s,
                                                const float* __restrict__ d,
                                                float* __restrict__ y1,
                                                float* __restrict__ y2) {
    const int wave = (blockIdx.x * blockDim.x + threadIdx.x) >> 5;
    const int lane = threadIdx.x & 31;
    const int n  = wave >> 6;          // batch
    const int it = (wave >> 2) & 15;   // i-tile (256/16)
    const int ft = wave & 3;           // f-tile (64/16)
    const int i0 = it * 16, f0 = ft * 16;

    const int l15   = lane & 15;
    const int khalf = (lane >> 4) * 2; // lanes 0-15: K 0,1 ; lanes 16-31: K 2,3
    const int arow  = i0 + l15;
    const int bcol  = f0 + l15;

    const float* sb = s + (size_t)n * V * V;
    const float* xb = x + (size_t)n * V * F;
    const float* db = d + n * V;

    v8f acc1 = {};
    v8f acc2 = {};
    for (int j0 = 0; j0 < V; j0 += 4) {
        const int ja = j0 + khalf, jb = ja + 1;
        const float a0 = -sb[arow * V + ja];       // contiguous pair -> b64 load
        const float a1 = -sb[arow * V + jb];
        v2f A   = {a0, a1};
        v2f Asq = {a0 * a0, a1 * a1};
        v2f B   = {xb[ja * F + bcol], xb[jb * F + bcol]};
        acc1 = __builtin_amdgcn_wmma_f32_16x16x4_f32(false, A,   false, B, (short)0, acc1, false, false);
        acc2 = __builtin_amdgcn_wmma_f32_16x16x4_f32(false, Asq, false, B, (short)0, acc2, false, false);
    }

    // D layout: lanes 0-15 -> rows m0..m0+7 with m0=0; lanes 16-31 -> m0=8.
    // Diagonal correction + store.
    const int m0 = (lane >> 4) * 8;
    float* y1b = y1 + (size_t)n * V * F;
    float* y2b = y2 + (size_t)n * V * F;
#pragma unroll
    for (int r = 0; r < 8; ++r) {
        const int row = i0 + m0 + r;
        const float xv  = xb[row * F + f0 + l15];
        const float c1  = db[row] - 1.0f;                 // l_t diag delta
        const float srr = sb[row * V + row];
        const float c2  = c1 * c1 - 2.0f * c1 * srr;      // (l_t^2) diag delta
        y1b[row * F + f0 + l15] = fmaf(c1, xv, acc1[r]);
        y2b[row * F + f0 + l15] = fmaf(c2, xv, acc2[r]);
    }
}

// gcn = relu( x @ (th0 - th2) + y1 @ th1 + y2 @ (2*th2) )
// Rows flattened over (n, v) -> 16384; one wave per 16x16 output tile.
__global__ __launch_bounds__(128) void cheb_out(const float* __restrict__ x,
                                                const float* __restrict__ theta,
                                                const float* __restrict__ y1,
                                                const float* __restrict__ y2,
                                                float* __restrict__ gcn) {
    const int wave = (blockIdx.x * blockDim.x + threadIdx.x) >> 5;
    const int lane = threadIdx.x & 31;
    const int rt = wave >> 2;          // [0,1024) row tiles
    const int ct = wave & 3;           // [0,4)   col tiles
    const int r0 = rt * 16, c0 = ct * 16;

    const int l15   = lane & 15;
    const int khalf = (lane >> 4) * 2;
    const int arow  = r0 + l15;
    const int bcol  = c0 + l15;

    const float* th0 = theta;
    const float* th1 = theta + F * C;
    const float* th2 = theta + 2 * F * C;

    v8f acc = {};
    for (int f0 = 0; f0 < F; f0 += 4) {
        const int ka = f0 + khalf, kb = ka + 1;
        v2f Ax  = {x [(size_t)arow * F + ka], x [(size_t)arow * F + kb]};
        v2f Ay1 = {y1[(size_t)arow * F + ka], y1[(size_t)arow * F + kb]};
        v2f Ay2 = {y2[(size_t)arow * F + ka], y2[(size_t)arow * F + kb]};
        v2f B02 = {th0[ka * C + bcol] - th2[ka * C + bcol],
                   th0[kb * C + bcol] - th2[kb * C + bcol]};
        v2f B1  = {th1[ka * C + bcol], th1[kb * C + bcol]};
        v2f B2  = {2.0f * th2[ka * C + bcol], 2.0f * th2[kb * C + bcol]};
        acc = __builtin_amdgcn_wmma_f32_16x16x4_f32(false, Ax,  false, B02, (short)0, acc, false, false);
        acc = __builtin_amdgcn_wmma_f32_16x16x4_f32(false, Ay1, false, B1,  (short)0, acc, false, false);
        acc = __builtin_amdgcn_wmma_f32_16x16x4_f32(false, Ay2, false, B2,  (short)0, acc, false, false);
    }

    const int m0 = (lane >> 4) * 8;
#pragma unroll
    for (int r = 0; r < 8; ++r) {
        const int row = r0 + m0 + r;
        gcn[(size_t)row * C + c0 + l15] = fmaxf(acc[r], 0.0f);
    }
}

extern "C" void kernel_launch(void* const* d_in, const int* in_sizes, int n_in,
                              void* d_out, int out_size, void* d_ws, size_t ws_size,
                              hipStream_t stream) {
    const float* x     = (const float*)d_in[0];
    const float* a     = (const float*)d_in[1];
    const float* theta = (const float*)d_in[2];

    float* gcn  = (float*)d_out;
    float* s    = gcn + GCN_ELEMS;
    float* loss = s + S_ELEMS;

    float* ws  = (float*)d_ws;
    float* dws = ws + WS_D;
    float* y1  = ws + WS_Y1;
    float* y2  = ws + WS_Y2;

    zero_loss<<<1, 1, 0, stream>>>(loss);
    graph_learn<<<NB, 256, 0, stream>>>(x, a, s, dws, loss);
    // 64 batches * 16 i-tiles * 4 f-tiles = 4096 waves, 4 waves/block
    lap_gemm<<<1024, 128, 0, stream>>>(x, s, dws, y1, y2);
    // 1024 row-tiles * 4 col-tiles = 4096 waves, 4 waves/block
    cheb_out<<<1024, 128, 0, stream>>>(x, theta, y1, y2, gcn);
}